// Haar_20907900797258
// MI455X (gfx1250) — compile-verified
//
#include <hip/hip_runtime.h>
#include <math.h>

typedef __attribute__((ext_vector_type(2))) float v2f;
typedef __attribute__((ext_vector_type(8))) float v8f;

#define BATCH      16
#define NIN        8192
#define NOUT       4096          // NIN/2
#define CH         256
#define M_ELEMS    (16.0*4096.0*256.0)
#define APPROX_N   (BATCH*NOUT*CH)   // 16777216
#define NBLOCKS    4096              // 16 batches * 64 n-tiles * 4 c-tiles

// One block: 64 output rows (n) x 64 channels (c) of one batch.
// Streams approx straight out, transposes details through a padded LDS tile,
// and accumulates sum(|d|) and sum(a) on the matrix pipe via WMMA f32 16x16x4
// with an all-ones B (D = A*ones == rowsum(A), replicated 16x over N).
__global__ __launch_bounds__(256) void haar_fwd_kernel(
    const float* __restrict__ x,
    float* __restrict__ approx,
    float* __restrict__ details,
    float* __restrict__ partials)
{
    __shared__ float tile[64][65];          // [c_local][n_local], stride 65 -> conflict free

    const float K = 0.70710678118654752440f;

    const int blk = blockIdx.x;
    const int b   = blk >> 8;               // 16 batches
    const int rem = blk & 255;
    const int n0  = (rem >> 2) << 6;        // 64 n-tiles of 64
    const int c0  = (rem & 3) << 6;         // 4  c-tiles of 64

    const int tid = threadIdx.x;
    const int j   = tid & 63;               // channel within tile
    const int ii  = tid >> 6;               // 0..3 row sub-index

    const float* xb = x       + (size_t)b * (NIN  * CH);
    float*       ab = approx  + (size_t)b * (NOUT * CH);
    float*       db = details + (size_t)b * (NOUT * CH);

    v8f accA = {};                           // WMMA C accumulator for sum(a)
    v8f accD = {};                           // WMMA C accumulator for sum(|d|)
    const v2f ones = {1.0f, 1.0f};

    float pa[2], pd[2];

    for (int t = 0; t < 8; ++t) {
        #pragma unroll
        for (int h = 0; h < 2; ++h) {
            const int i   = t * 8 + h * 4 + ii;       // 0..63 n_local
            const int row = n0 + i;                   // output row (pair index)
            const float* pe = xb + (size_t)(2 * row) * CH + c0 + j;
            // prefetch next row-pair (16 KB ahead) into cache
            if (h == 0) __builtin_prefetch(pe + 2 * 8 * CH, 0, 0);
            const float e = pe[0];
            const float o = pe[CH];
            const float a = (e + o) * K;
            const float d = (e - o) * K;
            ab[(size_t)row * CH + c0 + j] = a;        // coalesced 128B/wave
            tile[j][i] = d;                           // stage transpose
            pa[h] = a;
            pd[h] = fabsf(d);
        }
        // Feed the matrix pipe: 64 values/wave per call, ones-B => row sums into C.
        const v2f va = {pa[0], pa[1]};
        const v2f vd = {pd[0], pd[1]};
        accA = __builtin_amdgcn_wmma_f32_16x16x4_f32(
                   false, va, false, ones, (short)0, accA, false, false);
        accD = __builtin_amdgcn_wmma_f32_16x16x4_f32(
                   false, vd, false, ones, (short)0, accD, false, false);
    }

    __syncthreads();

    // Write transposed details: lanes sweep n_local -> coalesced 128B/wave.
    {
        const int i2 = tid & 63;                      // n_local
        const int jj = tid >> 6;                      // 0..3
        #pragma unroll
        for (int t = 0; t < 16; ++t) {
            const int jc = jj + t * 4;                // c_local
            db[(size_t)(c0 + jc) * 4096 + n0 + i2] = tile[jc][i2];
        }
    }
    __syncthreads();

    // Fold the per-lane C matrices, then block-reduce. Each distinct row-sum is
    // replicated 16x across N inside D, so scale by 1/16 at the end.
    float sA = 0.f, sD = 0.f;
    #pragma unroll
    for (int r = 0; r < 8; ++r) { sA += accA[r]; sD += accD[r]; }

    float* red = &tile[0][0];                         // reuse LDS: 512 floats
    red[tid]       = sA;
    red[256 + tid] = sD;
    __syncthreads();
    for (int s = 128; s > 0; s >>= 1) {
        if (tid < s) {
            red[tid]       += red[tid + s];
            red[256 + tid] += red[256 + tid + s];
        }
        __syncthreads();
    }
    if (tid == 0) {
        partials[2 * blk]     = red[0]   * 0.0625f;   // sum(a)   for this tile
        partials[2 * blk + 1] = red[256] * 0.0625f;   // sum(|d|) for this tile
    }
}

// Deterministic final reduction of 4096 block partials -> scalar r.
__global__ __launch_bounds__(256) void haar_finalize_kernel(
    const float* __restrict__ partials, float* __restrict__ rout)
{
    __shared__ float sA[256];
    __shared__ float sD[256];
    float a = 0.f, d = 0.f;
    for (int i = threadIdx.x; i < NBLOCKS; i += 256) {
        a += partials[2 * i];
        d += partials[2 * i + 1];
    }
    sA[threadIdx.x] = a;
    sD[threadIdx.x] = d;
    __syncthreads();
    for (int s = 128; s > 0; s >>= 1) {
        if (threadIdx.x < s) {
            sA[threadIdx.x] += sA[threadIdx.x + s];
            sD[threadIdx.x] += sD[threadIdx.x + s];
        }
        __syncthreads();
    }
    if (threadIdx.x == 0) {
        const double invs = 0.70710678118654752440;
        const double sumA = (double)sA[0];
        const double sumD = (double)sD[0];
        // rd = 0.01 * mean|details|
        const double rd = 0.01 * (sumD / M_ELEMS);
        // mean(approx)-mean(x) = sumA*(1-invsqrt2)/M   (since sum(x)=sumA*sqrt2)
        const double rc = 0.01 * fabs(sumA * (1.0 - invs) / M_ELEMS);
        rout[0] = (float)(rd + rc);
    }
}

extern "C" void kernel_launch(void* const* d_in, const int* in_sizes, int n_in,
                              void* d_out, int out_size, void* d_ws, size_t ws_size,
                              hipStream_t stream) {
    const float* x   = (const float*)d_in[0];
    float* out       = (float*)d_out;
    float* approx    = out;                       // 16*4096*256 floats
    float* r         = out + APPROX_N;            // 1 float
    float* details   = out + APPROX_N + 1;        // 16*4096*256 floats
    float* partials  = (float*)d_ws;              // 2*NBLOCKS floats (32 KB)

    haar_fwd_kernel<<<NBLOCKS, 256, 0, stream>>>(x, approx, details, partials);
    haar_finalize_kernel<<<1, 256, 0, stream>>>(partials, r);
}